// GraphNorm_9139690406327
// MI455X (gfx1250) — compile-verified
//
#include <hip/hip_runtime.h>
#include <hip/hip_bf16.h>
#include <stdint.h>

#define DD       128         // feature dim
#define GG       4096        // segments
#define CHUNK    48          // rows staged per LDS buffer
#define LDSTRIDE 144         // padded floats/row: 2*144 % 64-banks = 32 -> conflict-free B loads
#define NTHR     256         // 8 wave32s
#define EPSF     1e-5f

typedef __attribute__((ext_vector_type(2))) float f32x2;
typedef __attribute__((ext_vector_type(8))) float f32x8;

__device__ __forceinline__ int lower_bound_i(const int* __restrict__ a, int n, int v) {
    int lo = 0, hi = n;
    while (lo < hi) {
        int mid = (lo + hi) >> 1;
        if (a[mid] < v) lo = mid + 1; else hi = mid;
    }
    return lo;
}

// CDNA5 async global->LDS copy, 16B per lane, tracked by ASYNCcnt.
__device__ __forceinline__ void async_ld16(uint32_t lds_off, const float* gaddr) {
    asm volatile("global_load_async_to_lds_b128 %0, %1, off"
                 :: "v"(lds_off), "v"(gaddr)
                 : "memory");
}

__global__ void __launch_bounds__(NTHR)
graphnorm_kernel(const float* __restrict__ feat,
                 const int*   __restrict__ seg,
                 int n,
                 const float* __restrict__ weight,
                 const float* __restrict__ bias,
                 const float* __restrict__ mscale,
                 float* __restrict__ out)
{
    __shared__ __align__(16) float sbuf[2][CHUNK * LDSTRIDE];  // 54 KB double buffer
    __shared__ __align__(16) float cA_s[DD];                   // per-column scale
    __shared__ __align__(16) float cB_s[DD];                   // per-column shift

    const int tid  = threadIdx.x;
    const int lane = tid & 31;
    const int w    = tid >> 5;         // wave id 0..7 ; wave owns cols [16w, 16w+16)
    const int g    = blockIdx.x;

    // segment row range (ids are sorted)
    const int r0 = lower_bound_i(seg, n, g);
    const int r1 = lower_bound_i(seg, n, g + 1);
    const int count = r1 - r0;
    if (count <= 0) return;

    const uint32_t lb0 = (uint32_t)(size_t)&sbuf[0][0];
    const uint32_t lb1 = (uint32_t)(size_t)&sbuf[1][0];

    // ---- async staging: thread (w=rowgroup, lane=col-quad) copies 16B pieces ----
    const int colOff = lane * 4;       // quad start column for staging
    auto issue = [&](int chunkStart, uint32_t ldsBase) {
        #pragma unroll
        for (int j = 0; j < 6; ++j) {
            const int ric = j * 8 + w;
            int gr = chunkStart + ric;
            gr = gr < r1 ? gr : (r1 - 1);            // clamp tail: never OOB
            const float* ga = feat + (size_t)gr * DD + colOff;
            const uint32_t lo = ldsBase + (uint32_t)((ric * LDSTRIDE + colOff) * 4);
            async_ld16(lo, ga);
        }
    };

    const int nchunks = (count + CHUNK - 1) / CHUNK;

    // ---- WMMA reduction state: ones(16x4) x B(4x16) accumulated into C --------
    f32x8 csum = {};                   // per-wave column sums  (16 cols)
    f32x8 csq  = {};                   // per-wave column sum-of-squares
    const f32x2 ones = {1.0f, 1.0f};   // A-matrix: all ones

    // B operand row indices for this lane (K placement per 16x4 f32 layout)
    const int kOffV0 = (lane < 16) ? 0 : 2;   // VGPR0 K
    const int kOffV1 = (lane < 16) ? 1 : 3;   // VGPR1 K
    const int myCol  = w * 16 + (lane & 15);  // N = lane&15

    // ---------------- Pass 1: Σf and Σf² via async staging + WMMA --------------
    issue(r0, lb0);
    for (int c = 0; c < nchunks; ++c) {
        const int b = c & 1;
        if (c + 1 < nchunks) {
            issue(r0 + (c + 1) * CHUNK, b ? lb0 : lb1);        // prefetch next chunk
            asm volatile("s_wait_asynccnt 0x6" ::: "memory");  // chunk c done, c+1 in flight
        } else {
            asm volatile("s_wait_asynccnt 0x0" ::: "memory");
        }
        __syncthreads();                                       // chunk c visible to all waves

        const int    base = r0 + c * CHUNK;
        const float* bp   = &sbuf[b][0];
        #pragma unroll 4
        for (int k0 = 0; k0 < CHUNK; k0 += 4) {
            const int rV0 = k0 + kOffV0;
            const int rV1 = k0 + kOffV1;
            float x0 = bp[rV0 * LDSTRIDE + myCol];
            float x1 = bp[rV1 * LDSTRIDE + myCol];
            x0 = (base + rV0 < r1) ? x0 : 0.0f;                // zero tail rows
            x1 = (base + rV1 < r1) ? x1 : 0.0f;
            f32x2 bv = {x0, x1};
            f32x2 bq = {x0 * x0, x1 * x1};
            // D[m,n] = Σ_k 1*B[k,n] + C : matrix-engine column reduction
            csum = __builtin_amdgcn_wmma_f32_16x16x4_f32(false, ones, false, bv,
                                                         (short)0, csum, false, false);
            csq  = __builtin_amdgcn_wmma_f32_16x16x4_f32(false, ones, false, bq,
                                                         (short)0, csq,  false, false);
        }
        __syncthreads();                                       // safe to overwrite buffer b
    }

    // ---------------- Stats + fold affine: out = a*f + b -----------------------
    // All D rows are identical; C[0] in lanes 0-15 = col sums for N=lane.
    if (lane < 16) {
        const float S  = csum[0];
        const float Q  = csq[0];
        const float inv = 1.0f / (float)count;
        const float m  = S * inv;
        const float e2 = Q * inv;
        const float sc = mscale[myCol];
        const float var = e2 - m * m * sc * (2.0f - sc);       // E[(f-μs)²]
        const float a   = weight[myCol] / sqrtf(var + EPSF);
        cA_s[myCol] = a;
        cB_s[myCol] = bias[myCol] - a * m * sc;
    }
    __syncthreads();

    // ---------------- Pass 2: elementwise FMA (features re-read hits L2) -------
    const float4 A = *reinterpret_cast<const float4*>(&cA_s[lane * 4]);
    const float4 B = *reinterpret_cast<const float4*>(&cB_s[lane * 4]);
    const float4* __restrict__ fin  = reinterpret_cast<const float4*>(feat);
    float4*       __restrict__ fout = reinterpret_cast<float4*>(out);
    for (int r = r0 + w; r < r1; r += 8) {
        const size_t idx = (size_t)r * (DD / 4) + lane;
        float4 v = fin[idx];
        v.x = fmaf(A.x, v.x, B.x);
        v.y = fmaf(A.y, v.y, B.y);
        v.z = fmaf(A.z, v.z, B.z);
        v.w = fmaf(A.w, v.w, B.w);
        fout[idx] = v;
    }
}

extern "C" void kernel_launch(void* const* d_in, const int* in_sizes, int n_in,
                              void* d_out, int out_size, void* d_ws, size_t ws_size,
                              hipStream_t stream) {
    const float* feat   = (const float*)d_in[0];
    const int*   seg    = (const int*)d_in[1];
    // d_in[2] = num_graphs (device scalar); G fixed at 4096 per reference
    const float* weight = (const float*)d_in[3];
    const float* bias   = (const float*)d_in[4];
    const float* mscale = (const float*)d_in[5];
    float*       out    = (float*)d_out;
    const int n = in_sizes[0] / DD;

    graphnorm_kernel<<<GG, NTHR, 0, stream>>>(feat, seg, n, weight, bias, mscale, out);
}